// SeedConstructor_27006754357626
// MI455X (gfx1250) — compile-verified
//
#include <hip/hip_runtime.h>
#include <math.h>

#define D_MODEL 192
#define D_IN    800
#define PTS_PER_WG 64
#define XSTRIDE 808      // padded LDS row stride (floats) for x[64][800]
#define SSTRIDE 200      // staging stride for seed/pos [64][192]
#define LN_EPS 1e-5f
#define NPOINTS 32768    // B(8) * K(4096)

typedef float v2f __attribute__((ext_vector_type(2)));
typedef float v8f __attribute__((ext_vector_type(8)));

// ---- bilinear sample of one pyramid level (NCHW, square HxW) ----
__device__ __forceinline__ void sample_level(const float* __restrict__ feat, int Wl,
                                             float cx, float cy, int c0, int cstep,
                                             float* dst) {
    float scale = (float)(Wl - 1) * (1.0f / 511.0f);   // align_corners mapping
    float ix = cx * scale, iy = cy * scale;
    float ix0f = floorf(ix), iy0f = floorf(iy);
    float wx = ix - ix0f, wy = iy - iy0f;
    int x0 = (int)ix0f; x0 = x0 < 0 ? 0 : (x0 > Wl - 1 ? Wl - 1 : x0);
    int y0 = (int)iy0f; y0 = y0 < 0 ? 0 : (y0 > Wl - 1 ? Wl - 1 : y0);
    int x1 = (int)ix0f + 1; x1 = x1 < 0 ? 0 : (x1 > Wl - 1 ? Wl - 1 : x1);
    int y1 = (int)iy0f + 1; y1 = y1 < 0 ? 0 : (y1 > Wl - 1 ? Wl - 1 : y1);
    float w00 = (1.0f - wx) * (1.0f - wy);
    float w01 = wx * (1.0f - wy);
    float w10 = (1.0f - wx) * wy;
    float w11 = wx * wy;
    size_t hw = (size_t)Wl * Wl;
    const float* b00 = feat + (size_t)y0 * Wl + x0;
    const float* b01 = feat + (size_t)y0 * Wl + x1;
    const float* b10 = feat + (size_t)y1 * Wl + x0;
    const float* b11 = feat + (size_t)y1 * Wl + x1;
    for (int c = c0; c < D_MODEL; c += cstep) {
        size_t off = (size_t)c * hw;
        dst[c] = b00[off] * w00 + b01[off] * w01 + b10[off] * w10 + b11[off] * w11;
    }
}

// ---- pre-kernel: transpose W matrices into K-major layout for B-fragments ----
__global__ void transpose_kernel(const float* __restrict__ Wseed_w,
                                 const float* __restrict__ Wpos_w,
                                 float* __restrict__ WtS, float* __restrict__ WtP) {
    int i = blockIdx.x * blockDim.x + threadIdx.x;
    if (i < D_MODEL * D_IN) {
        int n = i / D_IN, k = i % D_IN;
        WtS[(size_t)k * D_MODEL + n] = Wseed_w[i];
    } else {
        int j = i - D_MODEL * D_IN;
        if (j < D_MODEL * 32) {
            int n = j / 32, k = j % 32;
            WtP[(size_t)k * D_MODEL + n] = Wpos_w[j];
        }
    }
}

__global__ __launch_bounds__(256) void seed_kernel(
    const float* __restrict__ coords,
    const float* __restrict__ L1, const float* __restrict__ L2,
    const float* __restrict__ L3, const float* __restrict__ L4,
    const float* __restrict__ Wseed_b, const float* __restrict__ ln_g,
    const float* __restrict__ ln_b,  const float* __restrict__ Wpos_b,
    const float* __restrict__ WtS,   const float* __restrict__ WtP,
    float* __restrict__ out_seed, float* __restrict__ out_pos,
    float* __restrict__ out_cg) {

    __shared__ float xlds[PTS_PER_WG * XSTRIDE];     // x matrix, later reused as staging
    __shared__ float s_mu[PTS_PER_WG], s_rs[PTS_PER_WG];

    const int tid = threadIdx.x;
    const int pbase = blockIdx.x * PTS_PER_WG;

    // ---------------- Phase 1: gather features + PE + center_grid ----------------
    {
        int pl  = tid >> 2;        // point within WG (0..63)
        int sub = tid & 3;         // 4 threads cooperate per point
        int pg  = pbase + pl;
        int b   = pg >> 12;        // / 4096
        float cx = coords[(size_t)pg * 2 + 0];
        float cy = coords[(size_t)pg * 2 + 1];
        float* row = &xlds[pl * XSTRIDE];
        sample_level(L1 + (size_t)b * D_MODEL * 128 * 128, 128, cx, cy, sub, 4, row);
        sample_level(L2 + (size_t)b * D_MODEL * 64  * 64,  64,  cx, cy, sub, 4, row + 192);
        sample_level(L3 + (size_t)b * D_MODEL * 32  * 32,  32,  cx, cy, sub, 4, row + 384);
        sample_level(L4 + (size_t)b * D_MODEL * 16  * 16,  16,  cx, cy, sub, 4, row + 576);
        if (sub == 0) {
            float xn = cx * (1.0f / 512.0f), yn = cy * (1.0f / 512.0f);
            float* pe = row + 768;
            float fpow = 1.0f;
            #pragma unroll
            for (int f = 0; f < 8; ++f) {
                float ax = 6.28318530717958647692f * fpow * xn;
                float ay = 6.28318530717958647692f * fpow * yn;
                pe[f]      = sinf(ax);
                pe[8 + f]  = cosf(ax);
                pe[16 + f] = sinf(ay);
                pe[24 + f] = cosf(ay);
                fpow *= 2.0f;
            }
            out_cg[(size_t)pg * 2 + 0] = cx * (2.0f / 511.0f) - 1.0f;
            out_cg[(size_t)pg * 2 + 1] = cy * (2.0f / 511.0f) - 1.0f;
        }
    }
    __syncthreads();

    // ---------------- Phase 2: WMMA GEMMs (seed: K=800, pos: K=32) ----------------
    const int lane   = tid & 31;
    const int w      = tid >> 5;        // wave id 0..7
    const int mt     = w & 3;           // M tile (16 points each)
    const int n0base = (w >> 2) * 96;   // N group: 6 tiles of 16 cols
    const int arow   = mt * 16 + (lane & 15);
    const int koff   = (lane >> 4) << 1;  // lanes 16-31 carry K+2,K+3 (ISA A-layout)
    const int ncol   = lane & 15;

    v8f accS[6], accP[6];
    #pragma unroll
    for (int t = 0; t < 6; ++t) {
        accS[t] = (v8f){0.f,0.f,0.f,0.f,0.f,0.f,0.f,0.f};
        accP[t] = (v8f){0.f,0.f,0.f,0.f,0.f,0.f,0.f,0.f};
    }

    for (int k0 = 0; k0 < D_IN; k0 += 4) {
        v2f a = *(const v2f*)&xlds[arow * XSTRIDE + k0 + koff];
        const float* bp = &WtS[(size_t)(k0 + koff) * D_MODEL + n0base + ncol];
        #pragma unroll
        for (int t = 0; t < 6; ++t) {
            v2f bf;
            bf.x = bp[t * 16];               // row k0+koff
            bf.y = bp[D_MODEL + t * 16];     // row k0+koff+1
            accS[t] = __builtin_amdgcn_wmma_f32_16x16x4_f32(
                false, a, false, bf, (short)0, accS[t], false, false);
        }
    }
    // pos GEMM over the 32 PE columns (x cols 768..799)
    for (int k0 = 0; k0 < 32; k0 += 4) {
        v2f a = *(const v2f*)&xlds[arow * XSTRIDE + 768 + k0 + koff];
        const float* bp = &WtP[(size_t)(k0 + koff) * D_MODEL + n0base + ncol];
        #pragma unroll
        for (int t = 0; t < 6; ++t) {
            v2f bf;
            bf.x = bp[t * 16];
            bf.y = bp[D_MODEL + t * 16];
            accP[t] = __builtin_amdgcn_wmma_f32_16x16x4_f32(
                false, a, false, bf, (short)0, accP[t], false, false);
        }
    }
    __syncthreads();   // done reading x from LDS

    // ---------------- Phase 3: stage seed (+bias), LayerNorm, write out ----------------
    {
        const int rbase = mt * 16 + ((lane >> 4) << 3);   // D layout: lanes 16-31 -> M+8
        #pragma unroll
        for (int t = 0; t < 6; ++t) {
            int c = n0base + t * 16 + (lane & 15);
            float bias = Wseed_b[c];
            #pragma unroll
            for (int v = 0; v < 8; ++v)
                xlds[(rbase + v) * SSTRIDE + c] = accS[t][v] + bias;
        }
    }
    __syncthreads();

    if (tid < PTS_PER_WG) {
        const float* r = &xlds[tid * SSTRIDE];
        float s = 0.f, s2 = 0.f;
        for (int c = 0; c < D_MODEL; ++c) { float v = r[c]; s += v; s2 += v * v; }
        float mu  = s * (1.0f / D_MODEL);
        float var = s2 * (1.0f / D_MODEL) - mu * mu;
        s_mu[tid] = mu;
        s_rs[tid] = rsqrtf(var + LN_EPS);
    }
    __syncthreads();

    for (int i = tid; i < PTS_PER_WG * D_MODEL; i += 256) {
        int r = i / D_MODEL, c = i - r * D_MODEL;
        float v = (xlds[r * SSTRIDE + c] - s_mu[r]) * s_rs[r] * ln_g[c] + ln_b[c];
        out_seed[(size_t)pbase * D_MODEL + i] = v;   // rows contiguous -> flat = base + i
    }
    __syncthreads();

    // stage pos (+bias) and write out
    {
        const int rbase = mt * 16 + ((lane >> 4) << 3);
        #pragma unroll
        for (int t = 0; t < 6; ++t) {
            int c = n0base + t * 16 + (lane & 15);
            float bias = Wpos_b[c];
            #pragma unroll
            for (int v = 0; v < 8; ++v)
                xlds[(rbase + v) * SSTRIDE + c] = accP[t][v] + bias;
        }
    }
    __syncthreads();
    for (int i = tid; i < PTS_PER_WG * D_MODEL; i += 256) {
        int r = i / D_MODEL, c = i - r * D_MODEL;
        out_pos[(size_t)pbase * D_MODEL + i] = xlds[r * SSTRIDE + c];
    }
}

extern "C" void kernel_launch(void* const* d_in, const int* in_sizes, int n_in,
                              void* d_out, int out_size, void* d_ws, size_t ws_size,
                              hipStream_t stream) {
    const float* coords  = (const float*)d_in[0];
    const float* L1      = (const float*)d_in[1];
    const float* L2      = (const float*)d_in[2];
    const float* L3      = (const float*)d_in[3];
    const float* L4      = (const float*)d_in[4];
    const float* Wseed_w = (const float*)d_in[5];
    const float* Wseed_b = (const float*)d_in[6];
    const float* ln_g    = (const float*)d_in[7];
    const float* ln_b    = (const float*)d_in[8];
    const float* Wpos_w  = (const float*)d_in[9];
    const float* Wpos_b  = (const float*)d_in[10];

    float* WtS = (float*)d_ws;                      // [800][192]
    float* WtP = WtS + (size_t)D_IN * D_MODEL;      // [32][192]

    float* out      = (float*)d_out;
    float* out_seed = out;
    float* out_pos  = out + (size_t)NPOINTS * D_MODEL;
    float* out_cg   = out_pos + (size_t)NPOINTS * D_MODEL;

    int tot = D_MODEL * D_IN + D_MODEL * 32;
    transpose_kernel<<<(tot + 255) / 256, 256, 0, stream>>>(Wseed_w, Wpos_w, WtS, WtP);
    seed_kernel<<<NPOINTS / PTS_PER_WG, 256, 0, stream>>>(
        coords, L1, L2, L3, L4, Wseed_b, ln_g, ln_b, Wpos_b, WtS, WtP,
        out_seed, out_pos, out_cg);
}